// TreeModel_56049323213613
// MI455X (gfx1250) — compile-verified
//
#include <hip/hip_runtime.h>
#include <math.h>

typedef __attribute__((ext_vector_type(2))) float v2f;
typedef __attribute__((ext_vector_type(8))) float v8f;

#define NROW   2048
#define NDIM   16
#define JSPLIT 8
#define NWAVES ((NROW / 16) * JSPLIT)   // 1024 waves
// workspace layout (floats)
#define WS_INVNORM  0            // 16 floats
#define WS_EXCEED   16
#define WS_EXTRA    17
#define WS_POSITIVE 18
#define WS_OV       32           // NWAVES floats
#define WS_SQ       (32 + NWAVES)

// ---------------------------------------------------------------------------
// Kernel 0: small O(N*D) stats. Deterministic LDS tree reductions, one block.
// per_dim_norm^2 = 2N*Sum(l^2) - 2*(Sum l)^2  -> f64 to survive cancellation.
// ---------------------------------------------------------------------------
__global__ void __launch_bounds__(256, 1)
k0_stats(const float* __restrict__ ce, const float* __restrict__ pl,
         const float* __restrict__ ph, const float* __restrict__ cnt,
         float* __restrict__ ws) {
  __shared__ float  spl[NDIM], sph[NDIM];
  __shared__ float  slr;
  __shared__ double rd1[256], rd2[256];
  __shared__ float  red[256];
  const int t = threadIdx.x;

  if (t < NDIM) { spl[t] = pl[t]; sph[t] = ph[t]; }
  __syncthreads();
  if (t == 0) {
    float mn = spl[0], mx = sph[0];
    for (int d = 1; d < NDIM; ++d) { mn = fminf(mn, spl[d]); mx = fmaxf(mx, sph[d]); }
    slr = mx - mn;                       // level_range
  }
  __syncthreads();

  // per-dim first/second moments of l (16 threads per dim)
  {
    const int d = t & 15, p = t >> 4;
    double s1 = 0.0, s2 = 0.0;
    for (int i = p; i < NROW; i += 16) {
      double x = (double)ce[i * 32 + d];
      s1 += x; s2 += x * x;
    }
    rd1[t] = s1; rd2[t] = s2;
  }
  __syncthreads();
  if (t < NDIM) {
    double S1 = 0.0, S2 = 0.0;
    for (int p = 0; p < 16; ++p) { S1 += rd1[p * 16 + t]; S2 += rd2[p * 16 + t]; }
    double n2 = 2.0 * (double)NROW * S2 - 2.0 * S1 * S1;
    if (n2 < 0.0) n2 = 0.0;
    float nrm = (float)sqrt(n2);
    ws[WS_INVNORM + t] = 1.0f / fmaxf(nrm, 1e-10f);
  }

  // scalar losses: exceed / extra / positive
  const float lr = slr;
  float ex = 0.f, et = 0.f, po = 0.f;
  for (int i = t; i < NROW; i += 256) {
    const float c = cnt[i];
#pragma unroll
    for (int d = 0; d < NDIM; ++d) {
      const float l = ce[i * 32 + d], h = ce[i * 32 + 16 + d];
      const float ed = h - l;
      ex += fmaxf(spl[d] + 1.0f - l, 0.f) + fmaxf(h - sph[d] - 1.0f, 0.f);
      const float num = fmaxf(ed / lr, 1e-10f);
      const float dv  = num / c;
      et += fminf(fabsf(logf(dv)), 100000.0f);   // clip(|log|, -2, 1e5) == min(|log|,1e5)
      po += fmaxf(expf(-ed), 1e-10f);
    }
  }
  red[t] = ex; __syncthreads();
  for (int s = 128; s >= 1; s >>= 1) { if (t < s) red[t] += red[t + s]; __syncthreads(); }
  if (t == 0) ws[WS_EXCEED] = red[0];
  __syncthreads();
  red[t] = et; __syncthreads();
  for (int s = 128; s >= 1; s >>= 1) { if (t < s) red[t] += red[t + s]; __syncthreads(); }
  if (t == 0) ws[WS_EXTRA] = red[0];
  __syncthreads();
  red[t] = po; __syncthreads();
  for (int s = 128; s >= 1; s >>= 1) { if (t < s) red[t] += red[t + s]; __syncthreads(); }
  if (t == 0) ws[WS_POSITIVE] = red[0];
}

// ---------------------------------------------------------------------------
// Kernel 1: pairwise N x N x D pass. Each wave owns a 16-row i-tile (registers)
// and one of 8 j-segments (bounds wave-scalar via readfirstlane -> SALU loop,
// EXEC stays all-ones for WMMA). Per step it builds a 16x4 A-fragment of
// values and reduces it with V_WMMA_F32_16X16X4_F32 against an all-ones B:
//   D = A*1 + C  =>  sum(all D comps) == 16 * sum(all A values).
// __launch_bounds__(128, 1): only 1024 waves exist in the grid, so trade
// occupancy for full register residency (kills the scratch spill/reload the
// default budget produced under unroll-by-2).
// ---------------------------------------------------------------------------
__global__ void __launch_bounds__(128, 1)
k1_pair(const float* __restrict__ ce, const float* __restrict__ om,
        float* __restrict__ ws) {
  const int lane  = threadIdx.x & 31;
  const int wib   = __builtin_amdgcn_readfirstlane((int)(threadIdx.x >> 5));
  const int wave  = (int)blockIdx.x * 4 + wib;       // blockDim.x == 128
  const int itile = wave >> 3;
  const int jseg  = wave & 7;
  const int i     = itile * 16 + (lane & 15);

  float li[NDIM], hi[NDIM], ig[NDIM], wn[NDIM];
  const float* ci = ce + i * 32;
#pragma unroll
  for (int d = 0; d < NDIM; ++d) {
    li[d] = ci[d];
    hi[d] = ci[16 + d];
    ig[d] = 1.0f / fmaxf(hi[d] - li[d], 1e-10f);   // 1/gap[i,d]
    wn[d] = ws[WS_INVNORM + d];                    // 1/per_dim_norm[d] (uniform -> SGPR)
  }

  v8f acc1 = {};   // overlap-weighted sums
  v8f acc2 = {};   // squared residuals
  const v2f ones = {1.0f, 1.0f};
  const int jhalf = (lane >> 4) << 1;              // lanes 16-31 take K=2,3
  const float* orow = om + (size_t)i * NROW;
  const int jbeg = jseg * (NROW / JSPLIT);         // scalar
  const int jend = jbeg + (NROW / JSPLIT);         // scalar

#pragma unroll 2
  for (int j0 = jbeg; j0 < jend; j0 += 4) {
    __builtin_prefetch(ce + (j0 + 8 + jhalf) * 32, 0, 0);  // global_prefetch_b8
    v2f aOv, aR;
#pragma unroll
    for (int v = 0; v < 2; ++v) {
      const int j = j0 + jhalf + v;
      const float* cj = ce + j * 32;
      float ov = 0.f, sim = 0.f;
#pragma unroll
      for (int d = 0; d < NDIM; ++d) {
        const float lj = cj[d], hj = cj[16 + d];
        const float o = fminf(hi[d], hj) - fmaxf(li[d], lj);
        ov  = fmaf(fmaxf(o, 0.f), ig[d], ov);
        sim = fmaf(fabsf(li[d] - lj), wn[d], sim);
      }
      ov = (i == j) ? 0.f : ov;                    // offdiag mask (v_cndmask)
      const float r = orow[j] - sim;
      aOv[v] = ov;
      aR[v]  = r * r;
    }
    acc1 = __builtin_amdgcn_wmma_f32_16x16x4_f32(false, aOv, false, ones,
                                                 (short)0, acc1, false, false);
    acc2 = __builtin_amdgcn_wmma_f32_16x16x4_f32(false, aR, false, ones,
                                                 (short)0, acc2, false, false);
  }

  float s1 = 0.f, s2 = 0.f;
#pragma unroll
  for (int c = 0; c < 8; ++c) { s1 += acc1[c]; s2 += acc2[c]; }
#pragma unroll
  for (int off = 16; off >= 1; off >>= 1) {
    s1 += __shfl_xor(s1, off, 32);
    s2 += __shfl_xor(s2, off, 32);
  }
  if (lane == 0) {
    ws[WS_OV + wave] = s1 * 0.0625f;   // /16: ones-B replicates row sums 16x
    ws[WS_SQ + wave] = s2 * 0.0625f;
  }
}

// ---------------------------------------------------------------------------
// Kernel 2: deterministic final combine.
// ---------------------------------------------------------------------------
__global__ void __launch_bounds__(32, 1)
k2_final(const float* __restrict__ ws, float* __restrict__ out) {
  if (blockIdx.x == 0 && threadIdx.x == 0) {
    float sOv = 0.f, sSq = 0.f;
    for (int w = 0; w < NWAVES; ++w) { sOv += ws[WS_OV + w]; sSq += ws[WS_SQ + w]; }
    out[0] = sqrtf(sSq)                 // 1.0 * loss_sim
           + 5.0f * ws[WS_EXTRA]
           + 2.0f * ws[WS_EXCEED]
           + 9.0f * sOv
           + 6.0f * ws[WS_POSITIVE];
  }
}

extern "C" void kernel_launch(void* const* d_in, const int* in_sizes, int n_in,
                              void* d_out, int out_size, void* d_ws, size_t ws_size,
                              hipStream_t stream) {
  (void)in_sizes; (void)n_in; (void)out_size; (void)ws_size;
  const float* ce  = (const float*)d_in[0];   // children_embedding (2048,32)
  const float* pl  = (const float*)d_in[1];   // parent_embedding_l (16,)
  const float* ph  = (const float*)d_in[2];   // parent_embedding_h (16,)
  const float* om  = (const float*)d_in[3];   // omega_normed (2048,2048)
  const float* cnt = (const float*)d_in[4];   // leaves_cnt (2048,1)
  float* ws  = (float*)d_ws;
  float* out = (float*)d_out;

  k0_stats<<<1, 256, 0, stream>>>(ce, pl, ph, cnt, ws);
  k1_pair<<<NWAVES / 4, 128, 0, stream>>>(ce, om, ws);   // 256 blocks x 4 waves
  k2_final<<<1, 32, 0, stream>>>(ws, out);
}